// transfer_matrix_27608049779337
// MI455X (gfx1250) — compile-verified
//
#include <hip/hip_runtime.h>
#include <stdint.h>

// Transfer-matrix method, restructured:
//   - per-wavelength constants (r_i, 2*pi*n_{i+1}/wl, |prod t_i|^2 * n_last)
//     precomputed once (kernel 1), SoA layout in d_ws for coalesced loads.
//   - main kernel tracks only the first row (a,b) of the S-matrix, division-free,
//     and processes TM_BU batch rows per thread: amortizes the 27 constant loads
//     4x and gives 4 independent dependency chains to hide v_sin/v_cos/v_exp
//     latency (this op is trans/VALU-bound; no K-dim => WMMA inapplicable).
// ws layout (floats, N = NUM wavelengths):
//   [0,7N)   r_re      [7N,14N)  r_im
//   [14N,20N) c_re     [20N,26N) c_im     [26N,27N) scale

#define TM_NMAT 8
#define TM_NL   (TM_NMAT - 2)   // 6 layers with thickness
#define TM_TPB  256
#define TM_BU   4               // batch rows per thread
#define TWO_PI_F 6.28318530717958647692f

typedef int tm_v2i __attribute__((vector_size(8)));
typedef __attribute__((address_space(1))) tm_v2i* tm_gv2i_p;   // global
typedef __attribute__((address_space(3))) tm_v2i* tm_lv2i_p;   // LDS

__global__ __launch_bounds__(TM_TPB)
void tm_precompute(const float* __restrict__ mat_n0,
                   const float* __restrict__ mat_k0,
                   const float* __restrict__ wl,
                   float* __restrict__ ws, int num) {
  int n = blockIdx.x * blockDim.x + threadIdx.x;
  if (n >= num) return;

  float nr[TM_NMAT], ni[TM_NMAT];
#pragma unroll
  for (int i = 0; i < TM_NMAT; ++i) {
    nr[i] = mat_n0[n * TM_NMAT + i];
    ni[i] = mat_k0[n * TM_NMAT + i];
  }

  float* rre = ws;
  float* rim = ws + 7 * num;
  float* cre = ws + 14 * num;
  float* cim = ws + 20 * num;
  float* scl = ws + 26 * num;

  float Pr = 1.0f, Pi = 0.0f;            // running product of t_i (complex)
#pragma unroll
  for (int i = 0; i < TM_NMAT - 1; ++i) {
    float sr = nr[i] + nr[i + 1], si = ni[i] + ni[i + 1];
    float dr = nr[i] - nr[i + 1], di = ni[i] - ni[i + 1];
    float inv = 1.0f / (sr * sr + si * si);
    rre[i * num + n] = (dr * sr + di * si) * inv;   // r = (ni-nj)/(ni+nj)
    rim[i * num + n] = (di * sr - dr * si) * inv;
    float tr = 2.0f * (nr[i] * sr + ni[i] * si) * inv;  // t = 2 ni/(ni+nj)
    float ti = 2.0f * (ni[i] * sr - nr[i] * si) * inv;
    float qr = Pr * tr - Pi * ti;
    float qi = Pr * ti + Pi * tr;
    Pr = qr; Pi = qi;
  }

  float f = TWO_PI_F / wl[n];
#pragma unroll
  for (int i = 0; i < TM_NL; ++i) {
    cre[i * num + n] = f * nr[i + 1];   // phi = c * x[b,i] (complex coeff)
    cim[i * num + n] = f * ni[i + 1];
  }
  scl[n] = (Pr * Pr + Pi * Pi) * nr[TM_NMAT - 1];
}

__global__ __launch_bounds__(TM_TPB)
void tm_main(const float* __restrict__ x,
             const float* __restrict__ ws,
             float* __restrict__ out, int num, int bs) {
  const int tile = blockIdx.x;                // n-tile
  const int b0   = blockIdx.y * TM_BU;        // first batch row of this block
  const int n    = tile * TM_TPB + threadIdx.x;
  const int nb   = (bs - b0 < TM_BU) ? (bs - b0) : TM_BU;

  __shared__ float s_x[TM_BU * TM_NL + 2];    // 24 floats used (+pad)
  const float* gx = x + (size_t)b0 * TM_NL;   // 4 consecutive rows = 96B contiguous

  // Stage x[b0..b0+nb) into LDS via the gfx1250 async global->LDS path:
  // one b64 transfer per lane (lanes 0..nb*3-1).
#if defined(__gfx1250__) && __has_builtin(__builtin_amdgcn_global_load_async_to_lds_b64)
  if ((int)threadIdx.x < nb * (TM_NL / 2)) {
    tm_gv2i_p g = (tm_gv2i_p)gx + threadIdx.x;       // strips const, adds AS1
    tm_lv2i_p l = (tm_lv2i_p)&s_x[0] + threadIdx.x;  // generic -> AS3
    __builtin_amdgcn_global_load_async_to_lds_b64(g, l, 0, 0);
  }
#if __has_builtin(__builtin_amdgcn_s_wait_asynccnt)
  __builtin_amdgcn_s_wait_asynccnt(0);
#else
  asm volatile("s_wait_asynccnt 0" ::: "memory");
#endif
#else
  if ((int)threadIdx.x < nb * TM_NL) s_x[threadIdx.x] = gx[threadIdx.x];
#endif
  __syncthreads();

  if (n >= num) return;

  const float* rre = ws;
  const float* rim = ws + 7 * num;
  const float* cre = ws + 14 * num;
  const float* cim = ws + 20 * num;
  const float* scl = ws + 26 * num;

  __builtin_prefetch(scl + n, 0, 0);    // gfx1250 global_prefetch_b8

  // TM_BU independent first-row S-matrix recursions (division-free).
  float ar[TM_BU], ai[TM_BU], br[TM_BU], bi[TM_BU];
#pragma unroll
  for (int u = 0; u < TM_BU; ++u) {
    ar[u] = 1.0f; ai[u] = 0.0f; br[u] = 0.0f; bi[u] = 0.0f;
  }

#pragma unroll
  for (int i = 0; i < TM_NMAT - 1; ++i) {
    const float rr = rre[i * num + n];
    const float ri = rim[i * num + n];
    float cr = 0.0f, ci = 0.0f;
    if (i < TM_NL) { cr = cre[i * num + n]; ci = cim[i * num + n]; }
#pragma unroll
    for (int u = 0; u < TM_BU; ++u) {
      // (a,b) <- (a + b*r, a*r + b)
      const float nar = ar[u] + (br[u] * rr - bi[u] * ri);
      const float nai = ai[u] + (br[u] * ri + bi[u] * rr);
      const float nbr = br[u] + (ar[u] * rr - ai[u] * ri);
      const float nbi = bi[u] + (ar[u] * ri + ai[u] * rr);
      if (i < TM_NL) {
        const float xv  = s_x[u * TM_NL + i];
        const float pr  = cr * xv;                 // Re(phi)
        const float pim = ci * xv;                 // Im(phi)
        float s, c;
        __sincosf(pr, &s, &c);
        const float ep = __expf(pim);              // exp(-i*phi) = ep*(c - i s)
        const float em = __expf(-pim);             // exp(+i*phi) = em*(c + i s)
        ar[u] = ep * (nar * c + nai * s);
        ai[u] = ep * (nai * c - nar * s);
        br[u] = em * (nbr * c - nbi * s);
        bi[u] = em * (nbi * c + nbr * s);
      } else {
        ar[u] = nar; ai[u] = nai; br[u] = nbr; bi[u] = nbi;
      }
    }
  }

  const float sv = scl[n];
#pragma unroll
  for (int u = 0; u < TM_BU; ++u) {
    if (u < nb) out[(size_t)(b0 + u) * num + n] = sv / (ar[u] * ar[u] + ai[u] * ai[u]);
  }
}

extern "C" void kernel_launch(void* const* d_in, const int* in_sizes, int n_in,
                              void* d_out, int out_size, void* d_ws, size_t ws_size,
                              hipStream_t stream) {
  (void)n_in; (void)out_size; (void)ws_size;
  const float* x  = (const float*)d_in[0];   // (BS, 6)
  const float* n0 = (const float*)d_in[1];   // (NUM, 8)
  const float* k0 = (const float*)d_in[2];   // (NUM, 8)
  const float* wl = (const float*)d_in[3];   // (NUM,)
  float* out = (float*)d_out;                // (BS, NUM)
  float* ws  = (float*)d_ws;                 // 27*NUM floats

  const int num = in_sizes[3];
  const int bs  = in_sizes[0] / TM_NL;
  const int tiles = (num + TM_TPB - 1) / TM_TPB;
  const int bblk  = (bs + TM_BU - 1) / TM_BU;

  tm_precompute<<<dim3(tiles), dim3(TM_TPB), 0, stream>>>(n0, k0, wl, ws, num);
  tm_main<<<dim3(tiles, bblk), dim3(TM_TPB), 0, stream>>>(x, ws, out, num, bs);
}